// SoftmaxRouter_49933289783890
// MI455X (gfx1250) — compile-verified
//
#include <hip/hip_runtime.h>
#include <hip/hip_bf16.h>

#define D_MODEL     2048
#define NUM_EXPERTS 64
#define TOP_K       4
#define N_TOKENS    16384
#define ROWS_PER_BLOCK 128   // 8 waves x 16 rows

#ifndef __has_builtin
#define __has_builtin(x) 0
#endif

#if defined(__AMDGCN__) && \
    __has_builtin(__builtin_amdgcn_global_load_async_to_lds_b128) && \
    __has_builtin(__builtin_amdgcn_global_load_async_to_lds_b32)
#define USE_ASYNC 1
#else
#define USE_ASYNC 0
#endif

#if USE_ASYNC
#define NBUF 2
#define KC   32              // K chunk (double buffered)
#define SU   36              // u tile stride: 16B aligned, conflict-free
#define SEK  36              // E^T tile stride
#else
#define NBUF 1
#define KC   64
#define SU   68
#define SEK  68
#endif

typedef __attribute__((ext_vector_type(2))) float v2f;
typedef __attribute__((ext_vector_type(4))) float v4f;
typedef __attribute__((ext_vector_type(8))) float v8f;

#if USE_ASYNC
// Builtin signatures (from hipcc diagnostics):
//   b128: (vector_size(16) int AS1*, vector_size(16) int AS3*, imm int, imm int)
//   b32 : (int AS1*, int AS3*, imm int, imm int)
typedef int gv4i __attribute__((vector_size(16)));
typedef __attribute__((address_space(1))) gv4i as1_v4i;
typedef __attribute__((address_space(3))) gv4i as3_v4i;
typedef __attribute__((address_space(1))) int  as1_i32;
typedef __attribute__((address_space(3))) int  as3_i32;

__device__ __forceinline__ as1_v4i* g128(const void* p) {
    return (as1_v4i*)(uintptr_t)p;
}
__device__ __forceinline__ as3_v4i* l128(const void* p) {
    // AS3 pointers are 32-bit LDS byte offsets == low 32 bits of generic ptr.
    return (as3_v4i*)(unsigned)(uintptr_t)p;
}
__device__ __forceinline__ as1_i32* g32(const void* p) {
    return (as1_i32*)(uintptr_t)p;
}
__device__ __forceinline__ as3_i32* l32(const void* p) {
    return (as3_i32*)(unsigned)(uintptr_t)p;
}
#endif

__global__ __launch_bounds__(256) void router_gemm_softmax_topk(
    const float* __restrict__ u, const float* __restrict__ E,
    const float* __restrict__ bias,
    int*   __restrict__ topk_i, float* __restrict__ topk_s,
    float* __restrict__ scores, float* __restrict__ colsum_partial)
{
    __shared__ __align__(16) float ulds[NBUF][ROWS_PER_BLOCK * SU];
    __shared__ __align__(16) float elds[NBUF][NUM_EXPERTS * SEK];
    __shared__ float wavecol[8][NUM_EXPERTS];

    const int tid  = threadIdx.x;
    const int wave = tid >> 5;
    const int lane = tid & 31;
    const int half = lane >> 4;      // which 16-lane half of the wave
    const int n16  = lane & 15;
    const int rowBase = blockIdx.x * ROWS_PER_BLOCK + wave * 16;
    const int row0    = blockIdx.x * ROWS_PER_BLOCK;

    v8f acc0 = {}, acc1 = {}, acc2 = {}, acc3 = {};

#if USE_ASYNC
    // Prologue: kick off async copies of the first chunk into buffer 0.
    {
        #pragma unroll
        for (int it = 0; it < 4; ++it) {          // u: 128 rows x 32 K, b128/lane
            int idx4 = tid + it * 256;            // 1024 float4s
            int r    = idx4 >> 3;                 // 0..127
            int k4   = (idx4 & 7) << 2;           // 0..28
            __builtin_amdgcn_global_load_async_to_lds_b128(
                g128(u + (row0 + r) * D_MODEL + k4),
                l128(&ulds[0][r * SU + k4]), 0, 0);
        }
        #pragma unroll
        for (int it = 0; it < 8; ++it) {          // E^T: 32 K x 64 n, b32 scatter
            int idx = tid + it * 256;             // 2048 floats
            int r   = idx >> 6;                   // k 0..31
            int c   = idx & 63;                   // expert
            __builtin_amdgcn_global_load_async_to_lds_b32(
                g32(E + r * NUM_EXPERTS + c),
                l32(&elds[0][c * SEK + r]), 0, 0);
        }
    }
#endif

    int buf = 0;
    for (int kc = 0; kc < D_MODEL; kc += KC) {
#if USE_ASYNC
        asm volatile("s_wait_asynccnt 0x0" ::: "memory");
        __syncthreads();   // buf ready for all waves; buf^1 free to overwrite
        if (kc + KC < D_MODEL) {
            const int kn = kc + KC;
            const int nb = buf ^ 1;
            #pragma unroll
            for (int it = 0; it < 4; ++it) {
                int idx4 = tid + it * 256;
                int r    = idx4 >> 3;
                int k4   = (idx4 & 7) << 2;
                __builtin_amdgcn_global_load_async_to_lds_b128(
                    g128(u + (row0 + r) * D_MODEL + kn + k4),
                    l128(&ulds[nb][r * SU + k4]), 0, 0);
            }
            #pragma unroll
            for (int it = 0; it < 8; ++it) {
                int idx = tid + it * 256;
                int r   = idx >> 6;
                int c   = idx & 63;
                __builtin_amdgcn_global_load_async_to_lds_b32(
                    g32(E + (kn + r) * NUM_EXPERTS + c),
                    l32(&elds[nb][c * SEK + r]), 0, 0);
            }
        }
#else
        // Synchronous staging fallback (KC=64 layout).
        #pragma unroll
        for (int it = 0; it < 8; ++it) {
            int idx4 = tid + it * 256;
            int r    = idx4 >> 4;
            int k4   = (idx4 & 15) << 2;
            v4f d = *(const v4f*)(u + (row0 + r) * D_MODEL + kc + k4);
            *(v4f*)(&ulds[0][r * SU + k4]) = d;
        }
        #pragma unroll
        for (int it = 0; it < 4; ++it) {
            int idx4 = tid + it * 256;
            int r    = idx4 >> 4;
            int c4   = (idx4 & 15) << 2;
            v4f d = *(const v4f*)(E + (kc + r) * NUM_EXPERTS + c4);
            elds[0][(c4 + 0) * SEK + r] = d.x;
            elds[0][(c4 + 1) * SEK + r] = d.y;
            elds[0][(c4 + 2) * SEK + r] = d.z;
            elds[0][(c4 + 3) * SEK + r] = d.w;
        }
        __syncthreads();
        if (kc + KC < D_MODEL) {
            int r    = tid >> 1;
            int koff = (tid & 1) << 5;
            __builtin_prefetch(u + (row0 + r) * D_MODEL + kc + KC + koff, 0, 3);
        }
#endif

        // WMMA: A 16x4 f32 (lane=M, half selects K pair), B 4x16 from elds[n][k].
        const float* ua = &ulds[buf][(wave * 16 + n16) * SU + 2 * half];
        const float* eb = &elds[buf][n16 * SEK + 2 * half];
        #pragma unroll
        for (int k = 0; k < KC; k += 4) {
            v2f a  = *(const v2f*)(ua + k);
            v2f b0 = *(const v2f*)(eb + 0 * 16 * SEK + k);
            v2f b1 = *(const v2f*)(eb + 1 * 16 * SEK + k);
            v2f b2 = *(const v2f*)(eb + 2 * 16 * SEK + k);
            v2f b3 = *(const v2f*)(eb + 3 * 16 * SEK + k);
            acc0 = __builtin_amdgcn_wmma_f32_16x16x4_f32(false, a, false, b0, (short)0, acc0, false, false);
            acc1 = __builtin_amdgcn_wmma_f32_16x16x4_f32(false, a, false, b1, (short)0, acc1, false, false);
            acc2 = __builtin_amdgcn_wmma_f32_16x16x4_f32(false, a, false, b2, (short)0, acc2, false, false);
            acc3 = __builtin_amdgcn_wmma_f32_16x16x4_f32(false, a, false, b3, (short)0, acc3, false, false);
        }
#if USE_ASYNC
        buf ^= 1;
#else
        __syncthreads();
#endif
    }

    // Epilogue: bias + softmax + top-4 + column sums. C VGPR j holds
    // row (rowBase + j + 8*half) striped across this half's 16 lanes.
    const float b0 = bias[ 0 + n16];
    const float b1 = bias[16 + n16];
    const float b2 = bias[32 + n16];
    const float b3 = bias[48 + n16];
    float c0 = 0.f, c1 = 0.f, c2 = 0.f, c3 = 0.f;

    #pragma unroll
    for (int j = 0; j < 8; ++j) {
        float v0 = acc0[j] + b0;
        float v1 = acc1[j] + b1;
        float v2 = acc2[j] + b2;
        float v3 = acc3[j] + b3;

        float mx = fmaxf(fmaxf(v0, v1), fmaxf(v2, v3));
        #pragma unroll
        for (int off = 1; off < 16; off <<= 1)
            mx = fmaxf(mx, __shfl_xor(mx, off, 16));

        float e0 = __expf(v0 - mx);
        float e1 = __expf(v1 - mx);
        float e2 = __expf(v2 - mx);
        float e3 = __expf(v3 - mx);
        float s = e0 + e1 + e2 + e3;
        #pragma unroll
        for (int off = 1; off < 16; off <<= 1)
            s += __shfl_xor(s, off, 16);
        float inv = 1.0f / s;
        float s0 = e0 * inv, s1 = e1 * inv, s2 = e2 * inv, s3 = e3 * inv;

        const int row = rowBase + j + 8 * half;
        float* sp = scores + row * NUM_EXPERTS + n16;
        sp[0] = s0; sp[16] = s1; sp[32] = s2; sp[48] = s3;

        c0 += s0; c1 += s1; c2 += s2; c3 += s3;

        // Top-4 via iterative argmax over this half's 16 lanes x 4 values.
        float vv0 = s0, vv1 = s1, vv2 = s2, vv3 = s3;
        #pragma unroll
        for (int p = 0; p < TOP_K; ++p) {
            float bv = vv0; int bi = n16;
            if (vv1 > bv) { bv = vv1; bi = 16 + n16; }
            if (vv2 > bv) { bv = vv2; bi = 32 + n16; }
            if (vv3 > bv) { bv = vv3; bi = 48 + n16; }
            #pragma unroll
            for (int off = 1; off < 16; off <<= 1) {
                float ov = __shfl_xor(bv, off, 16);
                int   oi = __shfl_xor(bi, off, 16);
                if (ov > bv || (ov == bv && oi < bi)) { bv = ov; bi = oi; }
            }
            if (n16 == 0) {
                topk_s[row * TOP_K + p] = bv;
                topk_i[row * TOP_K + p] = bi;
            }
            if ((bi & 15) == n16) {   // owner retires the winner (scores > 0)
                if      (bi < 16) vv0 = -1.0f;
                else if (bi < 32) vv1 = -1.0f;
                else if (bi < 48) vv2 = -1.0f;
                else              vv3 = -1.0f;
            }
        }
    }

    // Deterministic column-sum reduction: halves -> wave slot -> block partial.
    c0 += __shfl_xor(c0, 16, 32);
    c1 += __shfl_xor(c1, 16, 32);
    c2 += __shfl_xor(c2, 16, 32);
    c3 += __shfl_xor(c3, 16, 32);
    if (half == 0) {
        wavecol[wave][ 0 + n16] = c0;
        wavecol[wave][16 + n16] = c1;
        wavecol[wave][32 + n16] = c2;
        wavecol[wave][48 + n16] = c3;
    }
    __syncthreads();
    if (tid < NUM_EXPERTS) {
        float s = 0.f;
        #pragma unroll
        for (int w = 0; w < 8; ++w) s += wavecol[w][tid];
        colsum_partial[blockIdx.x * NUM_EXPERTS + tid] = s;
    }
}

__global__ __launch_bounds__(64) void router_aux_loss(
    const float* __restrict__ colsum_partial, float* __restrict__ aux)
{
    __shared__ float red[NUM_EXPERTS];
    const int e = threadIdx.x;  // 64 threads
    float s = 0.f;
    for (int b = 0; b < N_TOKENS / ROWS_PER_BLOCK; ++b)
        s += colsum_partial[b * NUM_EXPERTS + e];
    float m = s * (1.0f / (float)N_TOKENS);
    red[e] = m * m;
    __syncthreads();
    if (e == 0) {
        float t = 0.f;
        #pragma unroll
        for (int i = 0; i < NUM_EXPERTS; ++i) t += red[i];
        *aux = t * (float)NUM_EXPERTS;
    }
}

extern "C" void kernel_launch(void* const* d_in, const int* in_sizes, int n_in,
                              void* d_out, int out_size, void* d_ws, size_t ws_size,
                              hipStream_t stream) {
    const float* u    = (const float*)d_in[0];
    const float* E    = (const float*)d_in[1];
    const float* bias = (const float*)d_in[2];

    float* out    = (float*)d_out;
    int*   topk_i = (int*)out;                                   // 16384*4 int32
    float* topk_s = out + N_TOKENS * TOP_K;                      // 16384*4 f32
    float* scores = out + 2 * N_TOKENS * TOP_K;                  // 16384*64 f32
    float* aux    = out + 2 * N_TOKENS * TOP_K + N_TOKENS * NUM_EXPERTS; // 1 f32

    float* colsum_partial = (float*)d_ws;  // 128 blocks x 64 experts

    dim3 grid(N_TOKENS / ROWS_PER_BLOCK);
    router_gemm_softmax_topk<<<grid, 256, 0, stream>>>(
        u, E, bias, topk_i, topk_s, scores, colsum_partial);
    router_aux_loss<<<1, 64, 0, stream>>>(colsum_partial, aux);
}